// GraphConstructionHingeEmbeddingLoss_47210280517640
// MI455X (gfx1250) — compile-verified
//
#include <hip/hip_runtime.h>
#include <math.h>

// CDNA5 / gfx1250: wave32, WMMA 16x16x4 f32.
typedef __attribute__((ext_vector_type(2))) float v2f;
typedef __attribute__((ext_vector_type(8))) float v8f;

#define PT_THLD 0.9f
#define NBLOCKS 1024

__device__ __forceinline__ float wave_reduce_add(float v) {
#pragma unroll
  for (int off = 16; off > 0; off >>= 1)
    v += __shfl_xor(v, off, 32);
  return v;
}

// One wave processes 16 edges per iteration.
// Lanes 0..15  : edge e, dims {4c+0, 4c+1} of delta for K-chunk c  (A/B layout low half)
// Lanes 16..31 : edge e, dims {4c+2, 4c+3} of delta for K-chunk c  (A/B layout high half)
// Three V_WMMA_F32_16X16X4_F32 accumulate Gram(Delta) = Delta * Delta^T over K=12;
// diag(Gram) = squared distances of the 16 edges.
__global__ __launch_bounds__(256) void hinge_partials(
    const float* __restrict__ x,
    const int*   __restrict__ pid,
    const float* __restrict__ pt,
    const int*   __restrict__ esrc,
    const int*   __restrict__ edst,
    int n_edges,
    float* __restrict__ partial /* [gridDim.x * 3] */) {
  __shared__ float s_attr[8], s_rep[8], s_cnt[8];

  const int lane = threadIdx.x & 31;
  const int wv   = threadIdx.x >> 5;
  const int hlf  = lane >> 4;   // which K-half of each chunk this lane supplies
  const int e16  = lane & 15;   // edge-in-tile this lane serves

  const int nTiles     = (n_edges + 15) >> 4;
  const int waveGlobal = blockIdx.x * 8 + wv;
  const int waveStride = gridDim.x * 8;

  float attr = 0.f, rep = 0.f, cnt = 0.f;

  for (int tile = waveGlobal; tile < nTiles; tile += waveStride) {
    const int  eIdx  = tile * 16 + e16;
    const bool valid = eIdx < n_edges;

    // Edge indices: streamed once over the whole run -> non-temporal so the
    // 51 MB index array doesn't evict the L2-resident gather targets.
    int s = 0, d = 0;
    if (valid) {
      s = __builtin_nontemporal_load(esrc + eIdx);
      d = __builtin_nontemporal_load(edst + eIdx);
    }

    // Gather only the 6 dims this lane owns (lane pair splits the 48B row).
    v2f a0 = {0.f, 0.f}, a1 = {0.f, 0.f}, a2 = {0.f, 0.f};
    if (valid) {
      const float* ps = x + (size_t)s * 12 + hlf * 2;
      const float* pd = x + (size_t)d * 12 + hlf * 2;
      v2f s0 = *(const v2f*)(ps + 0);
      v2f s1 = *(const v2f*)(ps + 4);
      v2f s2 = *(const v2f*)(ps + 8);
      v2f d0 = *(const v2f*)(pd + 0);
      v2f d1 = *(const v2f*)(pd + 4);
      v2f d2 = *(const v2f*)(pd + 8);
      a0 = s0 - d0;
      a1 = s1 - d1;
      a2 = s2 - d2;
    }

    // A (16x4) and B = A^T (4x16) have identical per-lane register images for
    // this symmetric product, so pass the same operand twice.
    // EXEC is all-1s here (no divergence at this point; tail lanes carry zeros).
    v8f acc = {0.f, 0.f, 0.f, 0.f, 0.f, 0.f, 0.f, 0.f};
    acc = __builtin_amdgcn_wmma_f32_16x16x4_f32(false, a0, false, a0, (short)0, acc, false, false);
    acc = __builtin_amdgcn_wmma_f32_16x16x4_f32(false, a1, false, a1, (short)0, acc, false, false);
    acc = __builtin_amdgcn_wmma_f32_16x16x4_f32(false, a2, false, a2, (short)0, acc, false, false);

    // Extract diag(acc): element (n,n) lives at (vgpr n, lane n) for n<8 and
    // (vgpr n-8, lane n+16) for n>=8. Select the component this lane could
    // contribute, then bpermute it to the owner lane (0..15).
    const int comp = (lane < 16) ? (lane & 7) : ((lane - 24) & 7);
    float dg = acc[0];
#pragma unroll
    for (int i = 1; i < 8; ++i) dg = (comp == i) ? acc[i] : dg;
    const int srcLane = (e16 < 8) ? e16 : (e16 + 16);
    const float dist2 = __shfl(dg, srcLane, 32);

    if (hlf == 0 && valid) {
      const int   pidS = pid[s];
      const int   pidD = pid[d];
      const float ptS  = pt[s];
      const bool  trueEdge = (pidS == pidD) && (pidS > 0);
      const bool  highPt   = trueEdge && (ptS > PT_THLD);
      const float dist     = sqrtf(dist2);
      if (highPt) { attr += dist; cnt += 1.f; }           // P_ATTR == 1
      if (!trueEdge) rep += fmaxf(1.f - dist, 0.f);       // P_REP == 1, R_EMB == 1
    }
  }

  attr = wave_reduce_add(attr);
  rep  = wave_reduce_add(rep);
  cnt  = wave_reduce_add(cnt);
  if (lane == 0) { s_attr[wv] = attr; s_rep[wv] = rep; s_cnt[wv] = cnt; }
  __syncthreads();
  if (threadIdx.x == 0) {
    float A = 0.f, R = 0.f, C = 0.f;
#pragma unroll
    for (int w = 0; w < 8; ++w) { A += s_attr[w]; R += s_rep[w]; C += s_cnt[w]; }
    partial[blockIdx.x * 3 + 0] = A;
    partial[blockIdx.x * 3 + 1] = R;
    partial[blockIdx.x * 3 + 2] = C;
  }
}

// Single-wave deterministic final reduction over the per-block partials.
__global__ void hinge_finalize(const float* __restrict__ partial, int nblocks,
                               float* __restrict__ out) {
  float a = 0.f, r = 0.f, c = 0.f;
  for (int i = threadIdx.x; i < nblocks; i += 32) {
    a += partial[3 * i + 0];
    r += partial[3 * i + 1];
    c += partial[3 * i + 2];
  }
  a = wave_reduce_add(a);
  r = wave_reduce_add(r);
  c = wave_reduce_add(c);
  if (threadIdx.x == 0) {
    const float norm = c + 1e-8f;
    out[0] = a / norm;
    out[1] = r / norm;
  }
}

extern "C" void kernel_launch(void* const* d_in, const int* in_sizes, int n_in,
                              void* d_out, int out_size, void* d_ws, size_t ws_size,
                              hipStream_t stream) {
  const float* x   = (const float*)d_in[0];   // [N_NODES, 12] f32
  const int*   pid = (const int*)d_in[1];     // [N_NODES] i32
  const float* pt  = (const float*)d_in[2];   // [N_NODES] f32
  const int*   ei  = (const int*)d_in[3];     // [2, N_EDGES] i32
  const int n_edges = in_sizes[3] / 2;

  float* out     = (float*)d_out;             // [2] f32: (attr, rep)
  float* partial = (float*)d_ws;              // NBLOCKS * 3 floats (12 KB)

  hinge_partials<<<NBLOCKS, 256, 0, stream>>>(x, pid, pt, ei, ei + n_edges,
                                              n_edges, partial);
  hinge_finalize<<<1, 32, 0, stream>>>(partial, NBLOCKS, out);
}